// HCGNNConv_52037823758818
// MI455X (gfx1250) — compile-verified
//
#include <hip/hip_runtime.h>
#include <hip/hip_bf16.h>
#include <math.h>

typedef __attribute__((ext_vector_type(2))) float v2f;
typedef __attribute__((ext_vector_type(8))) float v8f;

#define IN_DIM    128
#define COM_DIM   128
#define N_ETYPE   5
#define N_HEADS   4
#define CH        (N_HEADS * N_ETYPE)   // 20 softmax channels
#define NEG_SLOPE 0.2f

// ---------------------------------------------------------------------------
// Kernel 1: WMMA f32 GEMM.
// mess[t][n][c] = mask[n][t] * (feat[n] @ W[t])[c] + com_signal[n][c]
// grid = (N/16, N_ETYPE), block = 256 (8 waves, wave w owns cols [16w,16w+16))
// A-frag (16x4):  lane L holds feat[node0+L%16][k + 2*(L/16) + {0,1}]
// B-frag (4x16):  lane L holds W[k + 2*(L/16) + {0,1}][col0 + L%16]
// D (16x16):      VGPR r / lane L -> node0 + r + 8*(L/16), col0 + L%16
// ---------------------------------------------------------------------------
__global__ void k_mess_wmma(const float* __restrict__ feat,
                            const float* __restrict__ com,
                            const float* __restrict__ mask,
                            const float* __restrict__ W,
                            float* __restrict__ mess,
                            int Nn) {
  const int lane  = threadIdx.x & 31;
  const int wave  = threadIdx.x >> 5;
  const int node0 = blockIdx.x * 16;
  const int t     = blockIdx.y;
  const int col0  = wave * 16;
  const int row   = lane & 15;
  const int khalf = lane >> 4;

  const float* A  = feat + (size_t)(node0 + row) * IN_DIM;
  const float* Wt = W + (size_t)t * IN_DIM * COM_DIM + col0 + row;

  v8f acc = {};
#pragma unroll
  for (int k = 0; k < IN_DIM; k += 4) {
    const int kk = k + 2 * khalf;
    v2f a, b;
    a.x = A[kk + 0];
    a.y = A[kk + 1];
    b.x = Wt[(size_t)(kk + 0) * COM_DIM];
    b.y = Wt[(size_t)(kk + 1) * COM_DIM];
    // D = A*B + C  (emits v_wmma_f32_16x16x4_f32)
    acc = __builtin_amdgcn_wmma_f32_16x16x4_f32(
        /*neg_a=*/false, a, /*neg_b=*/false, b,
        /*c_mod=*/(short)0, acc, /*reuse_a=*/false, /*reuse_b=*/false);
  }

  const int col = col0 + row;
#pragma unroll
  for (int r = 0; r < 8; ++r) {
    const int node = node0 + r + 8 * khalf;
    const float mk = mask[node * N_ETYPE + t];
    const float cs = com[(size_t)node * COM_DIM + col];
    mess[((size_t)t * Nn + node) * COM_DIM + col] = mk * acc[r] + cs;
  }
}

// ---------------------------------------------------------------------------
// Kernel 2: el[n,h,t] = dot(mess[t,n,:], attn_l[t,h,:]); er likewise.
// One wave per (n,t); lane holds 4 consecutive floats; butterfly reduce.
// ---------------------------------------------------------------------------
__global__ void k_attn(const float* __restrict__ mess,
                       const float* __restrict__ attn_l,
                       const float* __restrict__ attn_r,
                       float* __restrict__ el, float* __restrict__ er, int Nn) {
  const int lane = threadIdx.x & 31;
  const int gw   = blockIdx.x * (blockDim.x >> 5) + (threadIdx.x >> 5);
  if (gw >= Nn * N_ETYPE) return;
  const int n = gw / N_ETYPE;
  const int t = gw - n * N_ETYPE;

  const float4 mv = *(const float4*)(mess + ((size_t)t * Nn + n) * COM_DIM + lane * 4);
  float acc[2 * N_HEADS];
#pragma unroll
  for (int h = 0; h < N_HEADS; ++h) {
    const float4 av = *(const float4*)(attn_l + (size_t)(t * N_HEADS + h) * COM_DIM + lane * 4);
    const float4 bv = *(const float4*)(attn_r + (size_t)(t * N_HEADS + h) * COM_DIM + lane * 4);
    acc[2 * h + 0] = mv.x * av.x + mv.y * av.y + mv.z * av.z + mv.w * av.w;
    acc[2 * h + 1] = mv.x * bv.x + mv.y * bv.y + mv.z * bv.z + mv.w * bv.w;
  }
#pragma unroll
  for (int off = 16; off > 0; off >>= 1)
#pragma unroll
    for (int i = 0; i < 2 * N_HEADS; ++i)
      acc[i] += __shfl_xor(acc[i], off, 32);

  if (lane == 0) {
#pragma unroll
    for (int h = 0; h < N_HEADS; ++h) {
      el[(size_t)n * CH + h * N_ETYPE + t] = acc[2 * h + 0];
      er[(size_t)n * CH + h * N_ETYPE + t] = acc[2 * h + 1];
    }
  }
}

__device__ __forceinline__ float leaky(float e) {
  return e >= 0.0f ? e : NEG_SLOPE * e;
}

// Sign-correct float atomic max (mx initialized to -inf).
__device__ __forceinline__ void atomicMaxF(float* addr, float val) {
  if (val >= 0.0f) atomicMax((int*)addr, __float_as_int(val));
  else             atomicMin((unsigned int*)addr, __float_as_uint(val));
}

// ---------------------------------------------------------------------------
// Kernel 3: segment max over dst for all 20 channels.
// ---------------------------------------------------------------------------
__global__ void k_edge_max(const float* __restrict__ el, const float* __restrict__ er,
                           const int* __restrict__ src, const int* __restrict__ dst,
                           float* __restrict__ mx, int M) {
  const int m = blockIdx.x * blockDim.x + threadIdx.x;
  if (m >= M) return;
  const int s = src[m], d = dst[m];
  const float* pl = el + (size_t)s * CH;
  const float* pr = er + (size_t)d * CH;
  float* pm = mx + (size_t)d * CH;
#pragma unroll
  for (int j = 0; j < CH; ++j) atomicMaxF(&pm[j], leaky(pl[j] + pr[j]));
}

// ---------------------------------------------------------------------------
// Kernel 4: segment sum of exp(e - mx[dst]) over dst for all 20 channels.
// ---------------------------------------------------------------------------
__global__ void k_edge_sum(const float* __restrict__ el, const float* __restrict__ er,
                           const int* __restrict__ src, const int* __restrict__ dst,
                           const float* __restrict__ mx, float* __restrict__ den, int M) {
  const int m = blockIdx.x * blockDim.x + threadIdx.x;
  if (m >= M) return;
  const int s = src[m], d = dst[m];
  const float* pl = el + (size_t)s * CH;
  const float* pr = er + (size_t)d * CH;
  const float* pm = mx + (size_t)d * CH;
  float* pd = den + (size_t)d * CH;
#pragma unroll
  for (int j = 0; j < CH; ++j)
    atomicAdd(&pd[j], __expf(leaky(pl[j] + pr[j]) - pm[j]));
}

// ---------------------------------------------------------------------------
// Kernel 5: one wave per edge. coeff = sum_h a[m,h,etype[m]];
// rst[dst] += mess[etype, src, :] * coeff  (128-wide atomic scatter).
// ---------------------------------------------------------------------------
__global__ void k_scatter(const float* __restrict__ mess,
                          const float* __restrict__ el, const float* __restrict__ er,
                          const float* __restrict__ mx, const float* __restrict__ den,
                          const int* __restrict__ src, const int* __restrict__ dst,
                          const int* __restrict__ etype,
                          float* __restrict__ rst, int Nn, int M) {
  const int lane = threadIdx.x & 31;
  const int m    = blockIdx.x * (blockDim.x >> 5) + (threadIdx.x >> 5);
  if (m >= M) return;
  const int s = src[m], d = dst[m], te = etype[m];

  float coeff = 0.0f;  // all lanes compute identically (broadcast loads)
#pragma unroll
  for (int h = 0; h < N_HEADS; ++h) {
    const int j = h * N_ETYPE + te;
    const float e = leaky(el[(size_t)s * CH + j] + er[(size_t)d * CH + j]);
    coeff += __expf(e - mx[(size_t)d * CH + j]) / den[(size_t)d * CH + j];
  }

  const float4 mv = *(const float4*)(mess + ((size_t)te * Nn + s) * COM_DIM + lane * 4);
  float* r = rst + (size_t)d * COM_DIM + lane * 4;
  atomicAdd(&r[0], mv.x * coeff);
  atomicAdd(&r[1], mv.y * coeff);
  atomicAdd(&r[2], mv.z * coeff);
  atomicAdd(&r[3], mv.w * coeff);
}

// ---------------------------------------------------------------------------
// Init + final ELU
// ---------------------------------------------------------------------------
__global__ void k_init(float* __restrict__ mx, float* __restrict__ den,
                       float* __restrict__ rst, int nCh, int nRst) {
  const int i = blockIdx.x * blockDim.x + threadIdx.x;
  if (i < nCh) { mx[i] = __int_as_float(0xff800000u); den[i] = 0.0f; }
  if (i < nRst) rst[i] = 0.0f;
}

__global__ void k_elu(const float* __restrict__ rst, float* __restrict__ out, int n) {
  const int i = blockIdx.x * blockDim.x + threadIdx.x;
  if (i < n) {
    const float v = rst[i];
    out[i] = v > 0.0f ? v : (__expf(v) - 1.0f);
  }
}

extern "C" void kernel_launch(void* const* d_in, const int* in_sizes, int n_in,
                              void* d_out, int out_size, void* d_ws, size_t ws_size,
                              hipStream_t stream) {
  const float* feat = (const float*)d_in[0];
  const float* com  = (const float*)d_in[1];
  const float* mask = (const float*)d_in[2];   // node_etype_collector (N,5)
  const float* W    = (const float*)d_in[3];   // multi_linear (5,128,128)
  const float* al   = (const float*)d_in[4];   // attn_l (5,4,128)
  const float* ar   = (const float*)d_in[5];   // attn_r (5,4,128)
  const int*   src  = (const int*)d_in[6];
  const int*   dst  = (const int*)d_in[7];
  const int*   ety  = (const int*)d_in[8];

  const int Nn = in_sizes[0] / IN_DIM;  // 40000
  const int M  = in_sizes[6];           // 800000

  // workspace carve-up (floats): mess | el | er | mx | den | rst
  float* ws   = (float*)d_ws;
  float* mess = ws;                                          // 5*N*128
  float* el   = mess + (size_t)N_ETYPE * Nn * COM_DIM;       // N*20
  float* er   = el + (size_t)Nn * CH;                        // N*20
  float* mx   = er + (size_t)Nn * CH;                        // N*20
  float* den  = mx + (size_t)Nn * CH;                        // N*20
  float* rst  = den + (size_t)Nn * CH;                       // N*128

  const int nRst = Nn * COM_DIM;

  k_init<<<(nRst + 255) / 256, 256, 0, stream>>>(mx, den, rst, Nn * CH, nRst);

  dim3 g1(Nn / 16, N_ETYPE);
  k_mess_wmma<<<g1, 256, 0, stream>>>(feat, com, mask, W, mess, Nn);

  const int waves2 = Nn * N_ETYPE;
  k_attn<<<(waves2 + 7) / 8, 256, 0, stream>>>(mess, al, ar, el, er, Nn);

  k_edge_max<<<(M + 255) / 256, 256, 0, stream>>>(el, er, src, dst, mx, M);
  k_edge_sum<<<(M + 255) / 256, 256, 0, stream>>>(el, er, src, dst, mx, den, M);
  k_scatter<<<(M + 7) / 8, 256, 0, stream>>>(mess, el, er, mx, den, src, dst, ety,
                                             rst, Nn, M);
  k_elu<<<(nRst + 255) / 256, 256, 0, stream>>>(rst, (float*)d_out, nRst);
}